// GCNModelAE_UN_25769804170
// MI455X (gfx1250) — compile-verified
//
#include <hip/hip_runtime.h>

// ---------------------------------------------------------------------------
// GCN autoencoder forward for MI455X (gfx1250, wave32, WMMA)
//   mu = spmm(A, relu(spmm(A, x@W1)) @ W2)
// GEMMs use V_WMMA_F32_16X16X4_F32 (keeps reference f32 precision).
// SpMM uses float4 gathers + global_atomic_add_f32 scatter; intermediates
// (51MB each) stay resident in the 192MB L2.
// ---------------------------------------------------------------------------

typedef __attribute__((ext_vector_type(2))) float v2f;
typedef __attribute__((ext_vector_type(8))) float v8f;

#define N_NODES 100000
#define NFEAT   256
#define NHID    128
#define NCLASS  64
#define N_EDGES 1600000

// ---- zero fill (float4 stores) --------------------------------------------
__global__ void zero_f32v4(float4* __restrict__ p, long n4) {
  long i = (long)blockIdx.x * blockDim.x + threadIdx.x;
  if (i < n4) p[i] = make_float4(0.f, 0.f, 0.f, 0.f);
}

// ---- WMMA f32 GEMM: C[M,NCOL] = op(A)[M,K] @ B[K,NCOL] --------------------
// One wave (32 lanes) per 16-row M tile; loops all NCOL/16 N-tiles reusing
// the A fragment. A fragment layout (ISA 7.12.2, 32-bit A 16x4):
//   lanes 0-15  : VGPR0=K0, VGPR1=K1 for row M=lane
//   lanes 16-31 : VGPR0=K2, VGPR1=K3 for row M=lane-16
// B fragment (4x16, row striped across lanes like C/D):
//   VGPR0: lanes0-15 row K0, lanes16-31 row K2; VGPR1: rows K1 / K3
// C/D: VGPR r -> M = (lane<16 ? r : 8+r), N = lane&15.
template<int K, int NCOL, bool RELU_A>
__global__ __launch_bounds__(32)
void wmma_gemm_f32(const float* __restrict__ A,
                   const float* __restrict__ B,
                   float* __restrict__ C) {
  constexpr int NT = NCOL / 16;
  const int mtile = blockIdx.x;
  const int lane  = threadIdx.x;
  const int half  = lane >> 4;    // 0: K pair {0,1}, 1: K pair {2,3}
  const int lm    = lane & 15;

  const float* arow = A + (long)(mtile * 16 + lm) * K;

  v8f acc[NT];
#pragma unroll
  for (int t = 0; t < NT; ++t) { v8f z = {}; acc[t] = z; }

  for (int k0 = 0; k0 < K; k0 += 4) {
    const int kk = k0 + half * 2;           // even -> 8B-aligned v2f load
    v2f a = *(const v2f*)(arow + kk);
    if (RELU_A) {
      a.x = fmaxf(a.x, 0.0f);
      a.y = fmaxf(a.y, 0.0f);
    }
#pragma unroll
    for (int t = 0; t < NT; ++t) {
      v2f b;
      b.x = B[(kk + 0) * NCOL + t * 16 + lm];
      b.y = B[(kk + 1) * NCOL + t * 16 + lm];
      acc[t] = __builtin_amdgcn_wmma_f32_16x16x4_f32(
          /*neg_a=*/false, a, /*neg_b=*/false, b,
          /*c_mod=*/(short)0, acc[t], /*reuse_a=*/false, /*reuse_b=*/false);
    }
  }

#pragma unroll
  for (int t = 0; t < NT; ++t) {
#pragma unroll
    for (int r = 0; r < 8; ++r) {
      const int mrow = mtile * 16 + half * 8 + r;
      C[(long)mrow * NCOL + t * 16 + lm] = acc[t][r];
    }
  }
}

// ---- COO SpMM scatter: out[dst] += val * dense[src] -----------------------
// One thread per (edge, 4-feature chunk): float4 gather (mostly L2 hits,
// intermediates fit in 192MB L2) + 4x global_atomic_add_f32.
template<int F>
__global__ __launch_bounds__(256)
void spmm_scatter(const float* __restrict__ vals,
                  const int*   __restrict__ src,
                  const int*   __restrict__ dst,
                  const float* __restrict__ dense,
                  float*       __restrict__ out,
                  int n_edges) {
  constexpr int Q = F / 4;
  const long idx = (long)blockIdx.x * blockDim.x + threadIdx.x;
  if (idx >= (long)n_edges * Q) return;
  const int e = (int)(idx / Q);
  const int q = (int)(idx - (long)e * Q);

  const float v = vals[e];
  const int   s = src[e];
  const int   d = dst[e];

  const float4 m = *(const float4*)(dense + (long)s * F + q * 4);
  float* o = out + (long)d * F + q * 4;
  atomicAdd(o + 0, v * m.x);
  atomicAdd(o + 1, v * m.y);
  atomicAdd(o + 2, v * m.z);
  atomicAdd(o + 3, v * m.w);
}

// ---------------------------------------------------------------------------
extern "C" void kernel_launch(void* const* d_in, const int* in_sizes, int n_in,
                              void* d_out, int out_size, void* d_ws, size_t ws_size,
                              hipStream_t stream) {
  (void)in_sizes; (void)n_in; (void)out_size; (void)ws_size;

  const float* x     = (const float*)d_in[0];   // [N, 256]
  const float* W1    = (const float*)d_in[1];   // [256, 128]
  const float* W2    = (const float*)d_in[2];   // [128, 64]
  const float* evals = (const float*)d_in[3];   // [E]
  const int*   esrc  = (const int*)  d_in[4];   // [E]
  const int*   edst  = (const int*)  d_in[5];   // [E]
  float*       outp  = (float*)d_out;           // [N, 64]

  float* sup = (float*)d_ws;                      // N*NHID (reused for support2)
  float* agg = sup + (size_t)N_NODES * NHID;      // N*NHID

  const int MTILES = N_NODES / 16;                // 6250, exact

  // 1) support1 = x @ W1
  wmma_gemm_f32<NFEAT, NHID, false><<<MTILES, 32, 0, stream>>>(x, W1, sup);

  // 2) agg = 0
  {
    long n4 = (long)N_NODES * NHID / 4;
    zero_f32v4<<<(int)((n4 + 255) / 256), 256, 0, stream>>>((float4*)agg, n4);
  }

  // 3) agg[dst] += val * support1[src]
  {
    long nthr = (long)N_EDGES * (NHID / 4);
    spmm_scatter<NHID><<<(int)((nthr + 255) / 256), 256, 0, stream>>>(
        evals, esrc, edst, sup, agg, N_EDGES);
  }

  // 4) support2 = relu(agg) @ W2   (ReLU fused into A loads)
  wmma_gemm_f32<NHID, NCLASS, true><<<MTILES, 32, 0, stream>>>(agg, W2, sup);

  // 5) out = 0
  {
    long n4 = (long)N_NODES * NCLASS / 4;
    zero_f32v4<<<(int)((n4 + 255) / 256), 256, 0, stream>>>((float4*)outp, n4);
  }

  // 6) out[dst] += val * support2[src]
  {
    long nthr = (long)N_EDGES * (NCLASS / 4);
    spmm_scatter<NCLASS><<<(int)((nthr + 255) / 256), 256, 0, stream>>>(
        evals, esrc, edst, sup, outp, N_EDGES);
  }
}